// GraphIsomorphism_62586263437744
// MI455X (gfx1250) — compile-verified
//
#include <hip/hip_runtime.h>

typedef __attribute__((ext_vector_type(2))) float v2f;
typedef __attribute__((ext_vector_type(8))) float v8f;

#define N_NODES 100000
#define N_EDGES 1600000
#define D 64

// Kernel 1: rep = epsilon * x   (rep lives in d_out's second half; atomics accumulate into it)
__global__ void gin_init_rep(const float* __restrict__ x, const float* __restrict__ eps,
                             float* __restrict__ rep) {
    int i = blockIdx.x * blockDim.x + threadIdx.x;      // float4 index, exact grid
    const float4* x4 = (const float4*)x;
    float4* r4 = (float4*)rep;
    float e = eps[0];
    float4 v = x4[i];
    v.x *= e; v.y *= e; v.z *= e; v.w *= e;
    r4[i] = v;
}

// Kernel 2: rep[src] += edge_val * x[dst]  -- 16 lanes per edge, float4 gather, f32 global atomics
__global__ void gin_scatter(const float* __restrict__ x, const int* __restrict__ esrc,
                            const int* __restrict__ edst, const float* __restrict__ eval_,
                            float* __restrict__ rep) {
    long tid = (long)blockIdx.x * blockDim.x + threadIdx.x;
    int e = (int)(tid >> 4);
    int c = ((int)tid & 15) << 2;                       // feature chunk of 4
    int src = esrc[e];
    int dst = edst[e];
    float w = eval_[e];
    float4 v = *(const float4*)(x + (long)dst * D + c); // L2-resident gather
    float* o = rep + (long)src * D + c;
    __hip_atomic_fetch_add(o + 0, w * v.x, __ATOMIC_RELAXED, __HIP_MEMORY_SCOPE_AGENT);
    __hip_atomic_fetch_add(o + 1, w * v.y, __ATOMIC_RELAXED, __HIP_MEMORY_SCOPE_AGENT);
    __hip_atomic_fetch_add(o + 2, w * v.z, __ATOMIC_RELAXED, __HIP_MEMORY_SCOPE_AGENT);
    __hip_atomic_fetch_add(o + 3, w * v.w, __ATOMIC_RELAXED, __HIP_MEMORY_SCOPE_AGENT);
}

// Kernel 3: out = rep @ W + bias via V_WMMA_F32_16X16X4_F32.
// One wave per 16x16 output tile; K=64 in 16 steps of 4.
__global__ void gin_gemm_wmma(const float* __restrict__ rep, const float* __restrict__ W,
                              const float* __restrict__ bias, float* __restrict__ out) {
    int lane = threadIdx.x & 31;
    int wib  = threadIdx.x >> 5;
    int wid  = blockIdx.x * 8 + wib;                    // 25000 waves exactly
    int tile_m = wid >> 2;                              // 0..6249
    int tile_n = (wid & 3) << 4;                        // 0,16,32,48
    int half = lane >> 4;                               // 0: K lo pair, 1: K hi pair
    int m    = lane & 15;

    // A operand: lane(half,m) supplies rep[tile_m*16+m][k + 2*half + {0,1}]
    const float* arow = rep + (long)(tile_m * 16 + m) * D + 2 * half;
    // B operand: lane(half,m) supplies W[k + 2*half + {0,1}][tile_n + m]
    int ncol = tile_n + m;

    v8f acc = {};
#pragma unroll
    for (int k = 0; k < D; k += 4) {
        v2f a, b;
        a.x = arow[k];
        a.y = arow[k + 1];
        b.x = W[(k + 2 * half) * D + ncol];
        b.y = W[(k + 2 * half + 1) * D + ncol];
        acc = __builtin_amdgcn_wmma_f32_16x16x4_f32(
            /*neg_a=*/false, a, /*neg_b=*/false, b,
            /*c_mod=*/(short)0, acc, /*reuse_a=*/false, /*reuse_b=*/false);
    }

    // D layout: VGPR r -> M = tile_m*16 + half*8 + r, N = tile_n + m
    float bv = bias[ncol];
    int mbase = tile_m * 16 + half * 8;
#pragma unroll
    for (int r = 0; r < 8; ++r)
        out[(long)(mbase + r) * D + ncol] = acc[r] + bv;
}

extern "C" void kernel_launch(void* const* d_in, const int* in_sizes, int n_in,
                              void* d_out, int out_size, void* d_ws, size_t ws_size,
                              hipStream_t stream) {
    const float* x    = (const float*)d_in[0];
    const int*   esrc = (const int*)d_in[1];
    const int*   edst = (const int*)d_in[2];
    const float* ev   = (const float*)d_in[3];
    const float* W    = (const float*)d_in[4];
    const float* eps  = (const float*)d_in[5];
    const float* bias = (const float*)d_in[6];

    float* out = (float*)d_out;                      // [N, 64] fp32
    float* rep = out + (size_t)N_NODES * D;          // second output, built in place

    // 6.4M floats / 4 per thread / 256 = 6250 blocks (exact)
    gin_init_rep<<<(N_NODES * D / 4) / 256, 256, 0, stream>>>(x, eps, rep);
    // 1.6M edges * 16 lanes / 256 = 100000 blocks (exact)
    gin_scatter<<<(N_EDGES * 16) / 256, 256, 0, stream>>>(x, esrc, edst, ev, rep);
    // 25000 waves / 8 per block = 3125 blocks (exact)
    gin_gemm_wmma<<<(N_NODES / 16 * 4) / 8, 256, 0, stream>>>(rep, W, bias, out);
}